// AudioEffectsChain_73160472920645
// MI455X (gfx1250) — compile-verified
//
#include <hip/hip_runtime.h>

// ---------------- problem constants (from reference) ----------------
#define SR_F     44100.0f
#define T_LEN    44100
#define B_N      8
// chorus: 2 voices, phase offsets 0 and 0.5 -> sin and -sin
#define CH_BASE  882          // int(20.0*44100/1000)
#define CH_RANGE 220.0f       // int(10.0*44.1*0.5) = 220
#define CH_MIX   0.3f
// flanger
#define FL_BASE  220          // int(5.0*44100/1000)
#define FL_RANGE 123.0f       // int(4.0*44.1*0.7) = 123
#define FL_MIX   0.3f
#define FL_FB    0.3f
#define FL_BUF   512
#define FL_MASK  511
// min flanger delay = 220-123 = 97  ->  chunks of <=96 steps are parallel-safe
#define CHUNK    96
#define NCHUNK   ((T_LEN + CHUNK - 1) / CHUNK)   // 460

#define TWO_PI_F 6.28318530717958647692f

// Low 32 bits of a generic pointer to __shared__ = byte offset in the wave's
// LDS allocation (addrspacecast AS3->AS0 keeps the AS3 value in the low dword).
__device__ __forceinline__ unsigned lds_byte_off(const void* p) {
    return (unsigned)(unsigned long long)p;
}

// gfx1250 async global->LDS copy, per-lane addresses, tracked by ASYNCcnt.
__device__ __forceinline__ void async_ld_b32(unsigned lds_byte, const float* gsrc) {
    asm volatile("global_load_async_to_lds_b32 %0, %1, off"
                 :: "v"(lds_byte), "v"((unsigned long long)gsrc)
                 : "memory");
}

__global__ __launch_bounds__(CHUNK) void
AudioEffectsChain_fused_kernel(const float* __restrict__ x, float* __restrict__ out) {
    const int b = blockIdx.x;     // one workgroup (3 wave32s) per batch row / WGP
    const int j = threadIdx.x;    // timestep within chunk, 0..95
    const float* __restrict__ xb = x   + (size_t)b * T_LEN;
    float* __restrict__       ob = out + (size_t)b * T_LEN;

    __shared__ float buf[FL_BUF];          // flanger delay line (circular)
    __shared__ float stage[2][3][CHUNK];   // double-buffered chorus gathers

    // reference starts with a zeroed delay line
    for (int i = j; i < FL_BUF; i += CHUNK) buf[i] = 0.0f;

    // Issue the 3 chorus gathers for timestep t into stage[slot][*][j].
    // Out-of-range sources are address-clamped; validity flags returned in regs.
    auto issue = [&](int slot, int t, int& s0o, int& s1o) {
        int tt = (t < T_LEN) ? t : (T_LEN - 1);
        float v  = (float)tt * (1.5f / SR_F);      // t*CH_RATE/SR
        float ph = v - floorf(v);                  // % 1.0
        float sv = __sinf(TWO_PI_F * ph);          // voice0; voice1 = -sv (phase+0.5)
        int d0 = CH_BASE + (int)truncf( sv * CH_RANGE);
        int d1 = CH_BASE + (int)truncf(-sv * CH_RANGE);
        d0 = min(max(d0, 1), 2047);
        d1 = min(max(d1, 1), 2047);
        int s0 = tt - d0, s1 = tt - d1;
        async_ld_b32(lds_byte_off(&stage[slot][0][j]), xb + tt);
        async_ld_b32(lds_byte_off(&stage[slot][1][j]), xb + (s0 > 0 ? s0 : 0));
        async_ld_b32(lds_byte_off(&stage[slot][2][j]), xb + (s1 > 0 ? s1 : 0));
        s0o = s0; s1o = s1;
    };

    int s0_cur, s1_cur, s0_nxt, s1_nxt;
    issue(0, j, s0_cur, s1_cur);       // prefetch chunk 0
    __syncthreads();                   // buf zero-init visible to all waves

    for (int k = 0; k < NCHUNK; ++k) {
        const int t   = k * CHUNK + j;
        const int cur = k & 1, nxt = cur ^ 1;

        // software pipeline: start next chunk's gathers (always exactly 3/lane
        // so ASYNCcnt bookkeeping is uniform; last-iter issues are harmless,
        // s_endpgm's implicit wait-idle retires them)
        issue(nxt, t + CHUNK, s0_nxt, s1_nxt);

        // 6 outstanding -> wait to <=3 retires current chunk's 3 (in-order)
        asm volatile("s_wait_asynccnt 0x3" ::: "memory");

        if (t < T_LEN) {
            // ----- chorus (from async-staged LDS) -----
            float xc = stage[cur][0][j];
            float g0 = (s0_cur >= 0) ? stage[cur][1][j] : 0.0f;
            float g1 = (s1_cur >= 0) ? stage[cur][2][j] : 0.0f;
            // y = x*(1-MIX) + (MIX/2)*(g0+g1)
            float yc = xc * (1.0f - CH_MIX) + (0.5f * CH_MIX) * (g0 + g1);

            // ----- flanger step (chunk-parallel: min delay 97 > CHUNK-1) -----
            float vf  = (float)t * (0.5f / SR_F);  // t*FL_RATE/SR
            float phf = vf - floorf(vf);
            float sf  = __sinf(TWO_PI_F * phf);
            int df = FL_BASE + (int)truncf(sf * FL_RANGE);
            df = min(max(df, 1), FL_BUF - 1);
            int wi = t & FL_MASK;
            int ri = (wi - df) & FL_MASK;
            float delayed = buf[ri];               // written >=97 steps ago
            ob[t] = yc * (1.0f - FL_MIX) + delayed * FL_MIX;
            buf[wi] = fmaf(delayed, FL_FB, yc);    // provably no in-chunk alias
        }
        s0_cur = s0_nxt; s1_cur = s1_nxt;
        __syncthreads();                           // publish chunk's buf writes
    }
}

extern "C" void kernel_launch(void* const* d_in, const int* in_sizes, int n_in,
                              void* d_out, int out_size, void* d_ws, size_t ws_size,
                              hipStream_t stream) {
    (void)in_sizes; (void)n_in; (void)d_ws; (void)ws_size; (void)out_size;
    const float* x = (const float*)d_in[0];
    float* out = (float*)d_out;
    AudioEffectsChain_fused_kernel<<<dim3(B_N), dim3(CHUNK), 0, stream>>>(x, out);
}